// Pool_12919261627034
// MI455X (gfx1250) — compile-verified
//
#include <hip/hip_runtime.h>

typedef __attribute__((ext_vector_type(2))) float v2f;
typedef __attribute__((ext_vector_type(4))) float v4f;
typedef __attribute__((ext_vector_type(8))) float v8f;

#define B_      8
#define N_      4096
#define D_      512
#define K_      2048
#define NCHUNK  32

// ---------- Kernel A: deterministic partial column sums (no float atomics) ----------
__global__ void colsum_partial(const float* __restrict__ h, float* __restrict__ partial) {
    int b = blockIdx.y;
    int c = blockIdx.x;                 // chunk of rows
    int d = threadIdx.x;                // 0..511
    const float* base = h + ((size_t)b * N_ + (size_t)c * (N_ / NCHUNK)) * D_ + d;
    float acc = 0.f;
#pragma unroll 4
    for (int n = 0; n < N_ / NCHUNK; ++n)
        acc += base[(size_t)n * D_];
    partial[((size_t)b * NCHUNK + c) * D_ + d] = acc;
}

__global__ void colsum_reduce(const float* __restrict__ partial, float* __restrict__ col_sum) {
    int b = blockIdx.x;
    int d = threadIdx.x;
    float acc = 0.f;
#pragma unroll
    for (int c = 0; c < NCHUNK; ++c)
        acc += partial[((size_t)b * NCHUNK + c) * D_ + d];
    col_sum[b * D_ + d] = acc;
}

// ---------- Kernel B: scores via V_WMMA_F32_16X16X4_F32 (full fp32 precision) ----------
// One wave per 16-row tile. B matrix = col_sum broadcast across all 16 columns,
// so every column of D holds the same 16 scores. EXEC is all-ones (32 threads,
// no divergence before the WMMAs).
__global__ void scores_wmma(const float* __restrict__ h, const float* __restrict__ col_sum,
                            float* __restrict__ scores) {
    int lane  = threadIdx.x;            // 0..31
    int b     = blockIdx.y;
    int m0    = blockIdx.x * 16;
    int row   = m0 + (lane & 15);
    int khalf = (lane >> 4) * 2;        // lanes 0-15 -> K {0,1}; lanes 16-31 -> K {2,3}
    const float* hrow = h + ((size_t)b * N_ + row) * D_;
    const float* cs   = col_sum + (size_t)b * D_;

    v8f acc0 = {};
    v8f acc1 = {};
    for (int k = 0; k < D_; k += 8) {
        v2f a0; a0.x = hrow[k + khalf];     a0.y = hrow[k + khalf + 1];
        v2f b0; b0.x = cs[k + khalf];       b0.y = cs[k + khalf + 1];
        acc0 = __builtin_amdgcn_wmma_f32_16x16x4_f32(false, a0, false, b0,
                                                     (short)0, acc0, false, false);
        v2f a1; a1.x = hrow[k + 4 + khalf]; a1.y = hrow[k + 4 + khalf + 1];
        v2f b1; b1.x = cs[k + 4 + khalf];   b1.y = cs[k + 4 + khalf + 1];
        acc1 = __builtin_amdgcn_wmma_f32_16x16x4_f32(false, a1, false, b1,
                                                     (short)0, acc1, false, false);
    }
    v8f acc = acc0 + acc1;
    // C/D layout: VGPR r, lanes 0-15 -> M=r, lanes 16-31 -> M=8+r; all N columns equal.
    if (lane == 0) {
#pragma unroll
        for (int r = 0; r < 8; ++r) scores[(size_t)b * N_ + m0 + r] = acc[r];
    } else if (lane == 16) {
#pragma unroll
        for (int r = 0; r < 8; ++r) scores[(size_t)b * N_ + m0 + 8 + r] = acc[r];
    }
}

// ---------- Kernel C: exact K-smallest selection (radix select + scan compaction) ----------
// Matches jax top_k(-scores) tie-break (lower index first); emits indices sorted ascending.
__global__ __launch_bounds__(1024) void topk_select(const float* __restrict__ scores,
                                                    int* __restrict__ idx) {
    __shared__ unsigned keys[N_];
    __shared__ unsigned sc[1024];
    __shared__ unsigned hist[256];
    __shared__ unsigned sh_prefix;
    __shared__ int      sh_remK;

    int b = blockIdx.x;
    int t = threadIdx.x;

    // Order-preserving uint transform (ascending float == ascending uint).
    for (int i = t; i < N_; i += 1024) {
        unsigned u = __float_as_uint(scores[(size_t)b * N_ + i]);
        u = (u & 0x80000000u) ? ~u : (u | 0x80000000u);
        keys[i] = u;
    }
    if (t == 0) { sh_prefix = 0u; sh_remK = K_ - 1; }
    __syncthreads();

    // MSB-first radix select of the (K-1)-th smallest key.
    for (int shift = 24; shift >= 0; shift -= 8) {
        if (t < 256) hist[t] = 0u;
        __syncthreads();
        unsigned prefix    = sh_prefix;
        unsigned high_mask = (shift == 24) ? 0u : (0xFFFFFFFFu << (shift + 8));
#pragma unroll
        for (int q = 0; q < 4; ++q) {
            unsigned u = keys[t * 4 + q];
            if ((u & high_mask) == (prefix & high_mask))
                atomicAdd(&hist[(u >> shift) & 0xFFu], 1u);
        }
        __syncthreads();
        if (t == 0) {
            int rem = sh_remK;
            unsigned cum = 0, bin = 0;
            for (int bb = 0; bb < 256; ++bb) {
                unsigned hc = hist[bb];
                if ((unsigned)rem < cum + hc) { bin = (unsigned)bb; break; }
                cum += hc;
            }
            sh_remK   = rem - (int)cum;
            sh_prefix = prefix | (bin << shift);
        }
        __syncthreads();
    }
    unsigned T    = sh_prefix;                 // exact K-th smallest key value
    unsigned need = (unsigned)sh_remK + 1u;    // how many ties at T to keep
    __syncthreads();

    // Per-thread (4 contiguous elems) lt/eq counts, packed block scan.
    unsigned lt = 0, eq = 0;
#pragma unroll
    for (int q = 0; q < 4; ++q) {
        unsigned u = keys[t * 4 + q];
        lt += (u < T);
        eq += (u == T);
    }
    unsigned packed = (lt << 16) | eq;
    sc[t] = packed;
    __syncthreads();
    for (int off = 1; off < 1024; off <<= 1) {
        unsigned v = (t >= off) ? sc[t - off] : 0u;
        __syncthreads();
        sc[t] += v;
        __syncthreads();
    }
    unsigned excl = sc[t] - packed;
    unsigned ltp = excl >> 16;
    unsigned eqp = excl & 0xFFFFu;

#pragma unroll
    for (int q = 0; q < 4; ++q) {
        int i = t * 4 + q;
        unsigned u = keys[i];
        if (u < T) {
            unsigned pos = ltp + (eqp < need ? eqp : need);
            idx[b * K_ + pos] = i;
            ltp++;
        } else if (u == T) {
            if (eqp < need) idx[b * K_ + (ltp + eqp)] = i;
            eqp++;
        }
    }
}

// ---------- Kernel D: bandwidth-bound row gather (the real cost: ~550 MB traffic) ----------
__global__ void gather_rows(const float* __restrict__ h, const float* __restrict__ adj,
                            const int* __restrict__ idx, float* __restrict__ out) {
    int b = blockIdx.y;
    int p = blockIdx.x;
    int t = threadIdx.x;                       // 0..255
    int src = idx[b * K_ + p];

    float* out_h   = out;
    float* out_adj = out + (size_t)B_ * K_ * D_;

    // adj row: 4096 floats, stream with non-temporal hints (read-once/write-once).
    const v4f* arow = (const v4f*)(adj + ((size_t)b * N_ + src) * N_);
    v4f*       orow = (v4f*)(out_adj + ((size_t)b * K_ + p) * N_);
#pragma unroll
    for (int c = t; c < N_ / 4; c += 256) {
        v4f v = __builtin_nontemporal_load(&arow[c]);
        __builtin_nontemporal_store(v, &orow[c]);
    }
    // h row: 512 floats (h is L2-resident; keep regular temporal hint).
    if (t < D_ / 4) {
        const v4f* hrow  = (const v4f*)(h + ((size_t)b * N_ + src) * D_);
        v4f*       ohrow = (v4f*)(out_h + ((size_t)b * K_ + p) * D_);
        ohrow[t] = hrow[t];
    }
}

extern "C" void kernel_launch(void* const* d_in, const int* in_sizes, int n_in,
                              void* d_out, int out_size, void* d_ws, size_t ws_size,
                              hipStream_t stream) {
    const float* h   = (const float*)d_in[0];
    const float* adj = (const float*)d_in[1];
    float* out = (float*)d_out;
    float* ws  = (float*)d_ws;

    // Workspace layout (floats): partial[8*32*512] | col_sum[8*512] | scores[8*4096] | idx[8*2048 ints]
    float* partial = ws;
    float* col_sum = ws + (size_t)B_ * NCHUNK * D_;
    float* scores  = col_sum + (size_t)B_ * D_;
    int*   idx     = (int*)(scores + (size_t)B_ * N_);

    colsum_partial<<<dim3(NCHUNK, B_), 512, 0, stream>>>(h, partial);
    colsum_reduce<<<B_, 512, 0, stream>>>(partial, col_sum);
    scores_wmma<<<dim3(N_ / 16, B_), 32, 0, stream>>>(h, col_sum, scores);
    topk_select<<<B_, 1024, 0, stream>>>(scores, idx);
    gather_rows<<<dim3(K_, B_), 256, 0, stream>>>(h, adj, idx, out);
}